// LocalMatchingLoss_23699629540072
// MI455X (gfx1250) — compile-verified
//
#include <hip/hip_runtime.h>

// Problem constants (match reference)
#define B_    24
#define ND_   256
#define D_    2048
#define CONF_THRESH 0.5f
#define EPSF 1e-12f
#define NT 256

typedef float v2f __attribute__((ext_vector_type(2)));
typedef float v8f __attribute__((ext_vector_type(8)));

// ---------------------------------------------------------------------------
// Kernel 1: per-row inverse L2 norm for geom and text features.
// One 256-thread block per row (2048 elems -> 8 per thread), LDS reduction.
// ---------------------------------------------------------------------------
__global__ void lml_row_rnorm(const float* __restrict__ geom,
                              const float* __restrict__ text,
                              float* __restrict__ rg,
                              float* __restrict__ rt) {
    int row    = blockIdx.x;              // 0 .. 2*B*ND-1
    int isText = row >= B_ * ND_;
    int r      = isText ? row - B_ * ND_ : row;
    const float* src = (isText ? text : geom) + (size_t)r * D_;

    float ss = 0.f;
    for (int k = threadIdx.x; k < D_; k += NT) {
        float x = src[k];
        ss += x * x;
    }
    __shared__ float sh[NT];
    sh[threadIdx.x] = ss;
    __syncthreads();
    for (int s = NT / 2; s > 0; s >>= 1) {
        if (threadIdx.x < s) sh[threadIdx.x] += sh[threadIdx.x + s];
        __syncthreads();
    }
    if (threadIdx.x == 0) {
        float n = fmaxf(sqrtf(sh[0]), EPSF);
        (isText ? rt : rg)[r] = 1.0f / n;
    }
}

// ---------------------------------------------------------------------------
// Kernel 2: cosine similarity matrix via V_WMMA_F32_16X16X4_F32.
//   dot[b][tm][gn] = <text_norm[b,tm], geom_norm[b,gn]>
// One wave computes a 16x64 output strip (one A tile x four B tiles, four
// f32 accumulators) so each A fragment load feeds 4 WMMAs: 5 b64 loads per
// 4 WMMAs instead of 8 -> ~40% less L2 traffic, higher wmma density.
//
// 32-bit A 16x4 layout (ISA 7.12.2): lanes 0-15 hold M=0..15 with VGPR0=K0,
// VGPR1=K1; lanes 16-31 hold K2/K3. B (4x16) mirrored with lane = N.
// Normalization is folded into the epilogue (features read raw).
// ---------------------------------------------------------------------------
__global__ void lml_cost_wmma(const float* __restrict__ geom,
                              const float* __restrict__ text,
                              const float* __restrict__ rg,
                              const float* __restrict__ rt,
                              float* __restrict__ dotm) {
    int b    = blockIdx.z;
    int gn0  = blockIdx.x * 64;   // geom strip (WMMA N, 4 tiles of 16)
    int tm0  = blockIdx.y * 16;   // text tile  (WMMA M)
    int lane = threadIdx.x;       // 0..31, full wave, EXEC all ones
    int half = lane >> 4;         // 0: K0/K1 pair, 1: K2/K3 pair
    int l16  = lane & 15;

    const float* arow = text + ((size_t)b * ND_ + (tm0 + l16)) * D_;
    const float* br0  = geom + ((size_t)b * ND_ + (gn0 +      l16)) * D_;
    const float* br1  = geom + ((size_t)b * ND_ + (gn0 + 16 + l16)) * D_;
    const float* br2  = geom + ((size_t)b * ND_ + (gn0 + 32 + l16)) * D_;
    const float* br3  = geom + ((size_t)b * ND_ + (gn0 + 48 + l16)) * D_;

    v8f acc0 = {0.f,0.f,0.f,0.f,0.f,0.f,0.f,0.f};
    v8f acc1 = {0.f,0.f,0.f,0.f,0.f,0.f,0.f,0.f};
    v8f acc2 = {0.f,0.f,0.f,0.f,0.f,0.f,0.f,0.f};
    v8f acc3 = {0.f,0.f,0.f,0.f,0.f,0.f,0.f,0.f};

#pragma unroll 2
    for (int k = 0; k < D_; k += 4) {
        int kk = k + 2 * half;
        v2f a, b0, b1, b2, b3;
        a[0]  = arow[kk]; a[1]  = arow[kk + 1];
        b0[0] = br0[kk];  b0[1] = br0[kk + 1];
        b1[0] = br1[kk];  b1[1] = br1[kk + 1];
        b2[0] = br2[kk];  b2[1] = br2[kk + 1];
        b3[0] = br3[kk];  b3[1] = br3[kk + 1];
        acc0 = __builtin_amdgcn_wmma_f32_16x16x4_f32(false, a, false, b0,
                                                     (short)0, acc0, false, false);
        acc1 = __builtin_amdgcn_wmma_f32_16x16x4_f32(false, a, false, b1,
                                                     (short)0, acc1, false, false);
        acc2 = __builtin_amdgcn_wmma_f32_16x16x4_f32(false, a, false, b2,
                                                     (short)0, acc2, false, false);
        acc3 = __builtin_amdgcn_wmma_f32_16x16x4_f32(false, a, false, b3,
                                                     (short)0, acc3, false, false);
    }

    // Epilogue: scale by 1/||t|| * 1/||g|| and store dot[text][geom].
    // C/D layout: VGPR r, lanes 0-15 -> M=r, lanes 16-31 -> M=r+8; N=l16.
    float rg0 = rg[b * ND_ + gn0 +      l16];
    float rg1 = rg[b * ND_ + gn0 + 16 + l16];
    float rg2 = rg[b * ND_ + gn0 + 32 + l16];
    float rg3 = rg[b * ND_ + gn0 + 48 + l16];
    float* dbase = dotm + (size_t)b * ND_ * ND_;
#pragma unroll
    for (int r = 0; r < 8; ++r) {
        int m = tm0 + r + 8 * half;                 // text row
        float rtm = rt[b * ND_ + m];
        float* drow = dbase + (size_t)m * ND_ + gn0 + l16;
        drow[0]  = acc0[r] * rtm * rg0;
        drow[16] = acc1[r] * rtm * rg1;
        drow[32] = acc2[r] * rtm * rg2;
        drow[48] = acc3[r] * rtm * rg3;
    }
}

// ---------------------------------------------------------------------------
// Kernel 3: Jonker-Volgenant shortest augmenting path LSA, one block / batch.
// Rows = active text features (conf > 0.5), cols = 256 geom rows.
// Column scans, argmin and dual updates are parallel over 256 threads;
// potentials kept in double (mirrors numpy float64).
// ---------------------------------------------------------------------------
__global__ void lml_solver(const float* __restrict__ dotm,
                           const float* __restrict__ conf,
                           float* __restrict__ contrib,
                           int* __restrict__ validb) {
    const int b = blockIdx.x;
    const int m = ND_;
    const int tid = threadIdx.x;

    __shared__ int    act[ND_];
    __shared__ int    sK;
    __shared__ double u[ND_ + 1], v[ND_ + 1], minv[ND_ + 1];
    __shared__ int    p[ND_ + 1], way[ND_ + 1];
    __shared__ int    used[ND_ + 1];
    __shared__ double redv[NT];
    __shared__ int    redi[NT];
    __shared__ float  redf[NT];
    __shared__ int    sj0;

    const float* confb = conf + b * ND_;

    if (tid == 0) {
        int K = 0;
        for (int i = 0; i < ND_; ++i)
            if (confb[i] > CONF_THRESH) act[K++] = i;
        sK = K;
        if (K == 0) { contrib[b] = 0.f; validb[b] = 0; }
    }
    __syncthreads();
    const int n = sK;
    if (n == 0) return;

    for (int j = tid; j <= m; j += NT) {
        u[j] = 0.0; v[j] = 0.0; p[j] = 0; way[j] = 0;
    }
    __syncthreads();

    const float* dotb = dotm + (size_t)b * ND_ * ND_;

    for (int i = 1; i <= n; ++i) {
        if (tid == 0) { p[0] = i; sj0 = 0; }
        for (int j = tid; j <= m; j += NT) { minv[j] = 1e300; used[j] = 0; }
        __syncthreads();

        int j1;
        while (true) {
            int j0 = sj0;
            if (tid == 0) used[j0] = 1;
            __syncthreads();

            int i0 = p[j0];
            const float* crow = dotb + (size_t)act[i0 - 1] * ND_;
            double ui0 = u[i0];

            // Each thread owns column j = tid+1 (covers 1..256).
            int j = tid + 1;
            double mval = 1e300;
            if (!used[j]) {
                double cand = (double)(1.0f - crow[j - 1]) - ui0 - v[j];
                if (cand < minv[j]) { minv[j] = cand; way[j] = j0; }
                mval = minv[j];
            }
            redv[tid] = mval; redi[tid] = j;
            __syncthreads();
            for (int s = NT / 2; s > 0; s >>= 1) {
                if (tid < s && redv[tid + s] < redv[tid]) {
                    redv[tid] = redv[tid + s];
                    redi[tid] = redi[tid + s];
                }
                __syncthreads();
            }
            double delta = redv[0];
            j1 = redi[0];

            // Dual update: used columns have distinct assigned rows -> no
            // write conflicts on u[].
            for (int jj = tid; jj <= m; jj += NT) {
                if (used[jj]) { u[p[jj]] += delta; v[jj] -= delta; }
                else          { minv[jj] -= delta; }
            }
            if (tid == 0) sj0 = j1;
            __syncthreads();
            if (p[j1] == 0) break;
        }

        // Augment along alternating path (cheap, serial).
        if (tid == 0) {
            int j0 = sj0;
            while (j0 != 0) {
                int jw = way[j0];
                p[j0] = p[jw];
                j0 = jw;
            }
        }
        __syncthreads();
    }

    // Loss contribution: sum over matched columns of w * (1 - dot).
    float wsum = 0.f;
    {
        int j = tid + 1;
        if (p[j] != 0) {
            int i = p[j] - 1;        // local text index
            int t = act[i];          // global text row
            float pc = 1.0f - dotb[(size_t)t * ND_ + (j - 1)];
            wsum = confb[t] * pc;
        }
    }
    redf[tid] = wsum;
    __syncthreads();
    for (int s = NT / 2; s > 0; s >>= 1) {
        if (tid < s) redf[tid] += redf[tid + s];
        __syncthreads();
    }
    if (tid == 0) {
        contrib[b] = redf[0] / (float)n;
        validb[b]  = 1;
    }
}

// ---------------------------------------------------------------------------
// Kernel 4: deterministic final reduction over batches.
// ---------------------------------------------------------------------------
__global__ void lml_finalize(const float* __restrict__ contrib,
                             const int* __restrict__ validb,
                             float* __restrict__ out) {
    float t = 0.f;
    int   c = 0;
    for (int b = 0; b < B_; ++b) {
        if (validb[b]) { t += contrib[b]; ++c; }
    }
    out[0] = (c > 0) ? (t / (float)c) : 0.f;
}

// ---------------------------------------------------------------------------
extern "C" void kernel_launch(void* const* d_in, const int* in_sizes, int n_in,
                              void* d_out, int out_size, void* d_ws, size_t ws_size,
                              hipStream_t stream) {
    const float* geom = (const float*)d_in[0];  // [B, ND, D]
    const float* text = (const float*)d_in[1];  // [B, ND, D]
    const float* conf = (const float*)d_in[2];  // [B, ND]
    float* out = (float*)d_out;                 // scalar f32

    // Workspace layout (all re-written every launch -> no stale state).
    float* contrib = (float*)d_ws;                  // 24 floats (pad to 32)
    int*   validb  = (int*)(contrib + 32);          // 24 ints  (pad to 32)
    float* rg      = (float*)(validb + 32);         // B*ND
    float* rt      = rg + B_ * ND_;                 // B*ND
    float* dotm    = rt + B_ * ND_;                 // B*ND*ND (~6.3 MB)

    // 1) inverse norms
    lml_row_rnorm<<<2 * B_ * ND_, NT, 0, stream>>>(geom, text, rg, rt);

    // 2) cosine-similarity matrices via f32 WMMA (one wave per 16x64 strip)
    dim3 cgrid(ND_ / 64, ND_ / 16, B_);
    lml_cost_wmma<<<cgrid, 32, 0, stream>>>(geom, text, rg, rt, dotm);

    // 3) per-batch linear sum assignment + weighted matched cost
    lml_solver<<<B_, NT, 0, stream>>>(dotm, conf, contrib, validb);

    // 4) deterministic scalar reduction
    lml_finalize<<<1, 1, 0, stream>>>(contrib, validb, out);
}